// Deform_conv2d_90125593739963
// MI455X (gfx1250) — compile-verified
//
#include <hip/hip_runtime.h>
#include <hip/hip_bf16.h>
#include <math.h>

typedef __attribute__((ext_vector_type(2))) float v2f;
typedef __attribute__((ext_vector_type(8))) float v8f;

typedef __attribute__((address_space(1))) int as1_int;   // global
typedef __attribute__((address_space(3))) int as3_int;   // LDS

#define HH 256
#define WW 256
#define CIN 64
#define COUT 64
#define ROWS 4                       // output rows per block in kernel 3

#if defined(__HIP_DEVICE_COMPILE__) && __has_builtin(__builtin_amdgcn_global_load_async_to_lds_b32) && __has_builtin(__builtin_amdgcn_s_wait_asynccnt)
#define USE_ASYNC_LDS 1
#else
#define USE_ASYNC_LDS 0
#endif

// ---------------------------------------------------------------------------
// Kernel 1: fused offset conv (5x5, 64->2, pad 2) + bilinear grid sample.
// Block = 256 threads = one 16x16 pixel tile of one image.
// ---------------------------------------------------------------------------
__global__ __launch_bounds__(256) void k_offset_sample(
    const float* __restrict__ x, const float* __restrict__ w_def,
    const float* __restrict__ b_def, float* __restrict__ xs)
{
    __shared__ float swd[2 * CIN * 25];   // 3200 floats: both offset-channel weights
    __shared__ float sx[4 * 20 * 20];     // 4-channel input chunk with 2-halo

    const int tid = threadIdx.x;
    const int n  = blockIdx.z;
    const int y0 = blockIdx.y * 16;
    const int x0 = blockIdx.x * 16;

    for (int i = tid; i < 2 * CIN * 25; i += 256) swd[i] = w_def[i];

    const int ty = tid >> 4, tx = tid & 15;
    const int h = y0 + ty, w = x0 + tx;

    const float* xn = x + (size_t)n * CIN * HH * WW;
    float acc0 = 0.f, acc1 = 0.f;

    for (int c0 = 0; c0 < CIN; c0 += 4) {
        __syncthreads();
        for (int i = tid; i < 1600; i += 256) {
            int c = i / 400, r = (i % 400) / 20, col = i % 20;
            int gy = y0 - 2 + r, gx = x0 - 2 + col;
            float v = 0.f;
            if (gy >= 0 && gy < HH && gx >= 0 && gx < WW)
                v = xn[(size_t)(c0 + c) * HH * WW + (size_t)gy * WW + gx];
            sx[(c * 20 + r) * 20 + col] = v;
        }
        __syncthreads();
        for (int c = 0; c < 4; ++c) {
            const float* wp0 = &swd[(c0 + c) * 25];
            const float* wp1 = &swd[CIN * 25 + (c0 + c) * 25];
            const float* sp  = &sx[c * 400];
            #pragma unroll
            for (int ky = 0; ky < 5; ++ky) {
                #pragma unroll
                for (int kx = 0; kx < 5; ++kx) {
                    float v = sp[(ty + ky) * 20 + tx + kx];
                    acc0 = fmaf(v, wp0[ky * 5 + kx], acc0);
                    acc1 = fmaf(v, wp1[ky * 5 + kx], acc1);
                }
            }
        }
    }

    // grid = off + stack([xx, yy]); xx varies with h, yy with w (indexing='ij')
    float gx = acc0 + b_def[0] + (-1.f + 2.f * (float)h / (float)(HH - 1));
    float gy = acc1 + b_def[1] + (-1.f + 2.f * (float)w / (float)(WW - 1));
    float ix = ((gx + 1.f) * (float)WW - 1.f) * 0.5f;
    float iy = ((gy + 1.f) * (float)HH - 1.f) * 0.5f;
    float x0f = floorf(ix), y0f = floorf(iy);
    float x1f = x0f + 1.f,  y1f = y0f + 1.f;
    float wa = (x1f - ix) * (y1f - iy);   // pairs with (x0,y0)
    float wb = (x1f - ix) * (iy - y0f);   // pairs with (x0,y1)
    float wc = (ix - x0f) * (y1f - iy);   // pairs with (x1,y0)
    float wd = (ix - x0f) * (iy - y0f);   // pairs with (x1,y1)
    bool vx0 = (x0f >= 0.f) && (x0f <= (float)(WW - 1));
    bool vx1 = (x1f >= 0.f) && (x1f <= (float)(WW - 1));
    bool vy0 = (y0f >= 0.f) && (y0f <= (float)(HH - 1));
    bool vy1 = (y1f >= 0.f) && (y1f <= (float)(HH - 1));
    int xi0 = (int)fminf(fmaxf(x0f, 0.f), (float)(WW - 1));
    int xi1 = (int)fminf(fmaxf(x1f, 0.f), (float)(WW - 1));
    int yi0 = (int)fminf(fmaxf(y0f, 0.f), (float)(HH - 1));
    int yi1 = (int)fminf(fmaxf(y1f, 0.f), (float)(HH - 1));
    float fa = (vx0 && vy0) ? wa : 0.f;
    float fb = (vx0 && vy1) ? wb : 0.f;
    float fc = (vx1 && vy0) ? wc : 0.f;
    float fd = (vx1 && vy1) ? wd : 0.f;
    size_t o00 = (size_t)yi0 * WW + xi0;
    size_t o01 = (size_t)yi1 * WW + xi0;
    size_t o10 = (size_t)yi0 * WW + xi1;
    size_t o11 = (size_t)yi1 * WW + xi1;

    float* xsn = xs + (size_t)n * CIN * HH * WW + (size_t)h * WW + w;
    for (int c = 0; c < CIN; ++c) {
        const float* pl = xn + (size_t)c * HH * WW;
        float v = pl[o00] * fa + pl[o01] * fb + pl[o10] * fc + pl[o11] * fd;
        xsn[(size_t)c * HH * WW] = v;
    }
}

// ---------------------------------------------------------------------------
// Kernel 2: repack w_conv [co][ci][3][3] -> [tap][ci/2][co][ci%2]
// so B-fragment loads are one coalesced b64 per lane.
// ---------------------------------------------------------------------------
__global__ __launch_bounds__(256) void k_repack(
    const float* __restrict__ w, float* __restrict__ wr2)
{
    int idx = blockIdx.x * blockDim.x + threadIdx.x;
    if (idx >= COUT * CIN * 9) return;
    int tap = idx % 9;
    int ci  = (idx / 9) % CIN;
    int co  = idx / (9 * CIN);
    wr2[((size_t)(tap * (CIN / 2) + (ci >> 1)) * COUT + co) * 2 + (ci & 1)] = w[idx];
}

// ---------------------------------------------------------------------------
// Kernel 3: 3x3 conv (64->64, pad 1) as implicit GEMM on V_WMMA_F32_16X16X4_F32.
// Block = 256 threads (8 waves). Tile: fixed n, 4 output rows x 32 pixels x 64 cout.
// LDS: 6 rows x 34 px x 64 ci (channel stride padded to 68 -> conflict-free
// ds_load_b64 A-fragments). Each wave: 16 px x 16 cout x 4 rows = 576 WMMAs,
// with each B-fragment loaded once per tap and reused for all 4 rows.
// LDS fill uses gfx1250 GLOBAL_LOAD_ASYNC_TO_LDS_B32 when available.
// ---------------------------------------------------------------------------
__global__ __launch_bounds__(256) void k_conv3x3_wmma(
    const float* __restrict__ xs, const float* __restrict__ wr2,
    const float* __restrict__ b_conv, float* __restrict__ out)
{
    __shared__ float tile[(ROWS + 2) * 34 * 68];   // [row][x][ci], ci-stride 68

    const int tid = threadIdx.x;
    const int n = blockIdx.z, h0 = blockIdx.y * ROWS, w0 = blockIdx.x * 32;

    for (int i = tid; i < (ROWS + 2) * 34 * CIN; i += 256) {
        int ci = i / ((ROWS + 2) * 34);
        int rem = i - ci * ((ROWS + 2) * 34);
        int row = rem / 34, xx = rem - row * 34;
        int gy = h0 - 1 + row, gx = w0 - 1 + xx;
        float* lp = &tile[(row * 34 + xx) * 68 + ci];
        if (gy >= 0 && gy < HH && gx >= 0 && gx < WW) {
            const float* gp = &xs[(((size_t)n * CIN + ci) * HH + gy) * WW + gx];
#if USE_ASYNC_LDS
            __builtin_amdgcn_global_load_async_to_lds_b32(
                (as1_int*)const_cast<float*>(gp), (as3_int*)lp, 0, 0);
#else
            *lp = *gp;
#endif
        } else {
            *lp = 0.f;
        }
    }
#if USE_ASYNC_LDS
    __builtin_amdgcn_s_wait_asynccnt(0);
#endif
    __syncthreads();

    const int lane = tid & 31, wid = tid >> 5;
    const int p = wid & 1;        // pixel sub-tile (0..1)
    const int q = wid >> 1;       // cout sub-tile (0..3)
    const int m = lane & 15, half = lane >> 4;
    const int co = q * 16 + m;    // this lane's N (cout) index

    v8f acc[ROWS];
    #pragma unroll
    for (int r = 0; r < ROWS; ++r) acc[r] = (v8f){};

    for (int dy = 0; dy < 3; ++dy) {
        #pragma unroll
        for (int dx = 0; dx < 3; ++dx) {
            const int tap = dy * 3 + dx;
            // B fragments for this tap: loaded once, reused for all ROWS rows.
            const float* brow = wr2 + ((size_t)(tap * (CIN / 2) + half) * COUT + co) * 2;
            v2f bf[16];
            #pragma unroll
            for (int k = 0; k < 16; ++k)
                bf[k] = *(const v2f*)(brow + (size_t)k * (COUT * 2));
            #pragma unroll
            for (int r = 0; r < ROWS; ++r) {
                const float* arow =
                    &tile[((dy + r) * 34 + p * 16 + m + dx) * 68 + 2 * half];
                #pragma unroll
                for (int k = 0; k < 16; ++k) {
                    v2f a = *(const v2f*)(arow + 4 * k);      // ds_load_b64
                    acc[r] = __builtin_amdgcn_wmma_f32_16x16x4_f32(
                                 false, a, false, bf[k], (short)0, acc[r],
                                 false, false);
                }
            }
        }
    }

    const float bias = b_conv[co];
    #pragma unroll
    for (int r = 0; r < ROWS; ++r) {
        #pragma unroll
        for (int v = 0; v < 8; ++v) {
            int mpix = v + half * 8;              // C/D layout: M = v (+8 upper half)
            int xw = w0 + p * 16 + mpix;
            out[(((size_t)n * COUT + co) * HH + (h0 + r)) * WW + xw] = acc[r][v] + bias;
        }
    }
}

// ---------------------------------------------------------------------------
extern "C" void kernel_launch(void* const* d_in, const int* in_sizes, int n_in,
                              void* d_out, int out_size, void* d_ws, size_t ws_size,
                              hipStream_t stream) {
    const float* x      = (const float*)d_in[0];
    const float* w_def  = (const float*)d_in[1];
    const float* b_def  = (const float*)d_in[2];
    const float* w_conv = (const float*)d_in[3];
    const float* b_conv = (const float*)d_in[4];
    float* out = (float*)d_out;

    const size_t xs_elems = (size_t)8 * CIN * HH * WW;       // 134 MB
    float* xs  = (float*)d_ws;
    float* wr2 = (float*)((char*)d_ws + xs_elems * sizeof(float));

    k_offset_sample<<<dim3(WW / 16, HH / 16, 8), 256, 0, stream>>>(x, w_def, b_def, xs);
    k_repack<<<(COUT * CIN * 9 + 255) / 256, 256, 0, stream>>>(w_conv, wr2);
    k_conv3x3_wmma<<<dim3(WW / 32, HH / ROWS, 8), 256, 0, stream>>>(xs, wr2, b_conv, out);
}